// GNN_63058709840617
// MI455X (gfx1250) — compile-verified
//
#include <hip/hip_runtime.h>

#define NN 100000
#define EE 640000
#define DD 128

typedef __attribute__((ext_vector_type(2))) float v2f;
typedef __attribute__((ext_vector_type(8))) float v8f;

__global__ void zero_f32(float* __restrict__ p, long long n) {
    long long i = (long long)blockIdx.x * blockDim.x + threadIdx.x;
    if (i < n) p[i] = 0.0f;
}

__global__ void degree_k(const long long* __restrict__ ei, float* __restrict__ cnt) {
    int e = blockIdx.x * blockDim.x + threadIdx.x;
    if (e < EE) atomicAdd(cnt + (int)ei[EE + e], 1.0f);
}

__global__ void invcnt_k(const float* __restrict__ cnt, float* __restrict__ inv) {
    int i = blockIdx.x * blockDim.x + threadIdx.x;
    if (i < NN) inv[i] = 1.0f / fmaxf(cnt[i], 1.0f);
}

// One wave (32 lanes) per edge; each lane moves a float4 chunk of the 128-wide
// feature row: gather from x[src], atomic-add into agg[dst]. Feature matrix is
// 51 MB -> L2-resident on MI455X (192 MB L2), so these atomics stay on-chip.
__global__ void scatter_k(const float* __restrict__ x, const long long* __restrict__ ei,
                          float* __restrict__ agg) {
    long long t = (long long)blockIdx.x * blockDim.x + threadIdx.x;
    int e = (int)(t >> 5);
    if (e >= EE) return;
    int c = ((int)t & 31) << 2;          // column chunk: 32 lanes * 4 floats = 128
    int s = (int)ei[e];                  // src
    int d = (int)ei[EE + e];             // dst
    const float4 v = *(const float4*)(x + (size_t)s * DD + c);
    float* a = agg + (size_t)d * DD + c;
    atomicAdd(a + 0, v.x);
    atomicAdd(a + 1, v.y);
    atomicAdd(a + 2, v.z);
    atomicAdd(a + 3, v.w);
}

// out[m][n] = relu?( (agg[m]/cnt[m]) . Wl[n] + x[m] . Wr[n] + b[n] )
// 8 waves per block; wave w computes the 16x16 tile (rows blockIdx.x*16..+15,
// cols w*16..+15) with V_WMMA_F32_16X16X4_F32 accumulating over K=128.
// Fragment layouts per CDNA5 ISA 7.12.2:
//   A 16x4 f32 : lane&15 = M, (lane>>4) selects K {0,1} vs {2,3} -> float2 load
//   B  4x16 f32: lane&15 = N, same K split; B[k][n] = W[n][k] -> contiguous float2
//   C/D 16x16  : VGPR r -> M = r + 8*(lane>>4), N = lane&15
template <int RELU>
__global__ __launch_bounds__(256) void sage_gemm_k(
    const float* __restrict__ agg, const float* __restrict__ inv,
    const float* __restrict__ x,
    const float* __restrict__ Wl, const float* __restrict__ bias,
    const float* __restrict__ Wr, float* __restrict__ out) {
    const int wave = threadIdx.x >> 5;   // 0..7 -> column tile
    const int lane = threadIdx.x & 31;
    const int half = lane >> 4;          // which K pair / which M half (epilogue)
    const int l16  = lane & 15;
    const int m    = blockIdx.x * 16 + l16;   // A-fragment row
    const int n    = wave * 16 + l16;         // B-fragment col == output col
    const float sc = inv[m];                  // fused mean normalization
    const float* arow  = agg + (size_t)m * DD;
    const float* xrow  = x   + (size_t)m * DD;
    const float* wlrow = Wl  + (size_t)n * DD;
    const float* wrrow = Wr  + (size_t)n * DD;

    v8f acc = {};
#pragma unroll
    for (int kk = 0; kk < DD; kk += 4) {
        const int k2 = kk + 2 * half;
        v2f a0 = *(const v2f*)(arow + k2);
        a0 *= sc;
        v2f b0 = *(const v2f*)(wlrow + k2);
        acc = __builtin_amdgcn_wmma_f32_16x16x4_f32(false, a0, false, b0,
                                                    (short)0, acc, false, false);
        v2f a1 = *(const v2f*)(xrow + k2);
        v2f b1 = *(const v2f*)(wrrow + k2);
        acc = __builtin_amdgcn_wmma_f32_16x16x4_f32(false, a1, false, b1,
                                                    (short)0, acc, false, false);
    }

    const float bn = bias[n];
    const int m0 = blockIdx.x * 16 + half * 8;
#pragma unroll
    for (int r = 0; r < 8; ++r) {
        float v = acc[r] + bn;
        if (RELU) v = fmaxf(v, 0.0f);
        out[(size_t)(m0 + r) * DD + n] = v;
    }
}

extern "C" void kernel_launch(void* const* d_in, const int* in_sizes, int n_in,
                              void* d_out, int out_size, void* d_ws, size_t ws_size,
                              hipStream_t stream) {
    const float*     x   = (const float*)d_in[0];
    const long long* ei  = (const long long*)d_in[1];
    const float*     W1l = (const float*)d_in[2];
    const float*     b1  = (const float*)d_in[3];
    const float*     W1r = (const float*)d_in[4];
    const float*     W2l = (const float*)d_in[5];
    const float*     b2  = (const float*)d_in[6];
    const float*     W2r = (const float*)d_in[7];
    const float*     W3l = (const float*)d_in[8];
    const float*     b3  = (const float*)d_in[9];
    const float*     W3r = (const float*)d_in[10];
    float* out = (float*)d_out;

    float* agg = (float*)d_ws;                       // N*D
    float* cnt = agg + (size_t)NN * DD;              // N
    float* inv = cnt + NN;                           // N
    float* h1  = inv + NN;                           // N*D
    float* h2  = h1 + (size_t)NN * DD;               // N*D

    const dim3 blk(256);
    const long long aggN = (long long)NN * DD;
    const int aggBlocks  = (int)((aggN + 255) / 256);
    const long long scT  = (long long)EE * 32;
    const int scBlocks   = (int)((scT + 255) / 256);
    const int gemmBlocks = NN / 16;                  // 6250

    // degree / inverse-count (edge-structure invariant across layers)
    zero_f32<<<(NN + 255) / 256, blk, 0, stream>>>(cnt, NN);
    degree_k<<<(EE + 255) / 256, blk, 0, stream>>>(ei, cnt);
    invcnt_k<<<(NN + 255) / 256, blk, 0, stream>>>(cnt, inv);

    // layer 1
    zero_f32<<<aggBlocks, blk, 0, stream>>>(agg, aggN);
    scatter_k<<<scBlocks, blk, 0, stream>>>(x, ei, agg);
    sage_gemm_k<1><<<gemmBlocks, blk, 0, stream>>>(agg, inv, x, W1l, b1, W1r, h1);

    // layer 2
    zero_f32<<<aggBlocks, blk, 0, stream>>>(agg, aggN);
    scatter_k<<<scBlocks, blk, 0, stream>>>(h1, ei, agg);
    sage_gemm_k<1><<<gemmBlocks, blk, 0, stream>>>(agg, inv, h1, W2l, b2, W2r, h2);

    // layer 3 (no ReLU) -> d_out
    zero_f32<<<aggBlocks, blk, 0, stream>>>(agg, aggN);
    scatter_k<<<scBlocks, blk, 0, stream>>>(h2, ei, agg);
    sage_gemm_k<0><<<gemmBlocks, blk, 0, stream>>>(agg, inv, h2, W3l, b3, W3r, out);
}